// BackgroundAlignmentModule_78838419685496
// MI455X (gfx1250) — compile-verified
//
#include <hip/hip_runtime.h>

// MI455X / gfx1250, wave32. Exact-f32 local-correlation-softmax using
// V_WMMA_F32_16X16X4_F32 (D = A(16x4 f32) x B(4x16 f32) + C(16x16 f32)).
//
// 4 ordered feature pairs, b=2, c=64, h=w=128, radius=2 (5x5 window).
// One wave32 per (pair, batch, row y, 16-pixel x-tile). For each of the 5 dy
// rows: 16x32 f32 GEMM slice (two 16x16 N-tiles, K=64 via 16 WMMA K=4 steps),
// accumulators round-trip through 2KB LDS to extract the 5 diagonals j=m+dx
// per pixel, then a fully-register softmax over the 25 window scores yields
// (E[dx], E[dy]).
//
// Out-of-image columns are handled by CLAMPING the load address (always
// in-bounds) instead of predicating the load: the resulting garbage D entries
// are exactly the ones replaced by NEG_INF in the diagonal extraction, so no
// exec-mask manipulation is needed around the B loads (round-1 version paid
// ~320 s_and_saveexec/s_or_exec blocks for this).

typedef __attribute__((ext_vector_type(2))) float v2f;
typedef __attribute__((ext_vector_type(8))) float v8f;

#define Wd   128
#define Hd   128
#define Cd   64
#define HWd  (128 * 128)
#define NEG_INF_F (-1000000000.0f)

__global__ __launch_bounds__(32) void
lcs_wmma_kernel(const float* __restrict__ f0,
                const float* __restrict__ f1,
                const float* __restrict__ f2,
                float* __restrict__ out)
{
    __shared__ float lds[512];   // two 16x16 f32 accumulator tiles

    const int idx = blockIdx.x;          // 4 * 2 * 128 * 8 = 8192 blocks
    const int xt  = idx & 7;             // x tile (16 px each)
    const int y   = (idx >> 3) & 127;    // pixel row
    const int bb  = (idx >> 10) & 1;     // batch
    const int p   = idx >> 11;           // pair id 0..3

    const float* fs;
    const float* ft;
    if (p == 0)      { fs = f0; ft = f2; }
    else if (p == 1) { fs = f1; ft = f2; }
    else if (p == 2) { fs = f2; ft = f0; }
    else             { fs = f2; ft = f1; }

    const int l    = threadIdx.x;        // 0..31
    const int m    = l & 15;             // pixel-in-tile (A rows / C rows / B cols)
    const int half = l >> 4;             // K-half selector per WMMA A/B layout
    const int x0   = xt * 16;

    // ---- A matrix: 16 pixels x 64 channels, as 16 chunks of K=4 ----
    // 32-bit A 16x4 layout: lanes 0-15 -> M=lane, VGPR0=K0, VGPR1=K1;
    //                       lanes 16-31 -> M=lane-16, VGPR0=K2, VGPR1=K3.
    const float* sbase = fs + (size_t)bb * Cd * HWd + (size_t)y * Wd + x0 + m;
    v2f a[16];
#pragma unroll
    for (int kc = 0; kc < 16; ++kc) {
        const int ch = kc * 4 + 2 * half;
        a[kc].x = sbase[(size_t)ch * HWd];
        a[kc].y = sbase[(size_t)(ch + 1) * HWd];
    }

    float corrv[25];
    const float* tbb = ft + (size_t)bb * Cd * HWd;

    // B column mapping: tile0 covers absolute x = x0-2 .. x0+13 (j = 0..15),
    //                   tile1 covers absolute x = x0+14 .. x0+29 (j = 16..31).
    // Clamp out-of-image columns: their D entries never survive (NEG_INF'd).
    const int xc0 = x0 - 2 + m;
    const int xc1 = x0 + 14 + m;
    const int xc0c = (xc0 < 0) ? 0 : xc0;                // xc0 <= 125 always
    const int xc1c = (xc1 > Wd - 1) ? (Wd - 1) : xc1;    // xc1 >= 14 always

#pragma unroll
    for (int dyi = 0; dyi < 5; ++dyi) {
        const int yy = y + dyi - 2;
        const bool rowvalid = (yy >= 0) && (yy < Hd);   // uniform across wave
        if (!rowvalid) {
#pragma unroll
            for (int dxi = 0; dxi < 5; ++dxi) corrv[dyi * 5 + dxi] = NEG_INF_F;
        } else {
            const float* trow = tbb + (size_t)yy * Wd;
            // Per-lane base pointers; channel offsets become load immediates.
            const float* p0 = trow + xc0c + (size_t)half * HWd;
            const float* p1 = trow + xc1c + (size_t)half * HWd;

            v8f acc0 = {0.f, 0.f, 0.f, 0.f, 0.f, 0.f, 0.f, 0.f};
            v8f acc1 = {0.f, 0.f, 0.f, 0.f, 0.f, 0.f, 0.f, 0.f};

#pragma unroll
            for (int kc = 0; kc < 16; ++kc) {
                // 32-bit B 4x16 layout (mirror of A): lane 0-15 -> N=lane,
                // VGPR selects K pair, lane-half selects K within pair.
                v2f b0, b1;
                b0.x = p0[(size_t)(kc * 4) * HWd];
                b0.y = p0[(size_t)(kc * 4 + 2) * HWd];
                b1.x = p1[(size_t)(kc * 4) * HWd];
                b1.y = p1[(size_t)(kc * 4 + 2) * HWd];

                acc0 = __builtin_amdgcn_wmma_f32_16x16x4_f32(
                    false, a[kc], false, b0, (short)0, acc0, false, false);
                acc1 = __builtin_amdgcn_wmma_f32_16x16x4_f32(
                    false, a[kc], false, b1, (short)0, acc1, false, false);
            }

            // ---- spill C tiles to LDS to extract diagonals per pixel ----
            // C layout: VGPR v, lane l -> M = v + 8*(l>>4), N = l&15.
#pragma unroll
            for (int v = 0; v < 8; ++v) {
                const int M = v + 8 * half;
                lds[M * 16 + m]       = acc0[v];
                lds[256 + M * 16 + m] = acc1[v];
            }
            __syncthreads();

#pragma unroll
            for (int dxi = 0; dxi < 5; ++dxi) {
                const int j  = m + dxi;             // 0..19 within 20-col span
                const float val = (j < 16) ? lds[m * 16 + j]
                                           : lds[256 + m * 16 + (j - 16)];
                const int xs = x0 + m + dxi - 2;
                const bool valid = (xs >= 0) && (xs < Wd);
                corrv[dyi * 5 + dxi] =
                    valid ? val * 0.125f : NEG_INF_F;   // 1/sqrt(64)
            }
            __syncthreads();
        }
    }

    // ---- softmax over 25 window positions + expected offset ----
    float mx = corrv[0];
#pragma unroll
    for (int k = 1; k < 25; ++k) mx = fmaxf(mx, corrv[k]);

    float s = 0.f, ex = 0.f, ey = 0.f;
#pragma unroll
    for (int k = 0; k < 25; ++k) {
        const float pr = __expf(corrv[k] - mx);
        s  += pr;
        ex += pr * (float)((k % 5) - 2);
        ey += pr * (float)((k / 5) - 2);
    }
    const float inv = 1.0f / s;

    if (l < 16) {
        // out shape [4, 2, 2, 128, 128]
        const size_t obase =
            (((size_t)p * 2 + bb) * 2) * HWd + (size_t)y * Wd + (x0 + m);
        out[obase]               = ex * inv;   // channel 0: E[dx]
        out[obase + (size_t)HWd] = ey * inv;   // channel 1: E[dy]
    }
}

extern "C" void kernel_launch(void* const* d_in, const int* in_sizes, int n_in,
                              void* d_out, int out_size, void* d_ws, size_t ws_size,
                              hipStream_t stream) {
    (void)in_sizes; (void)n_in; (void)out_size; (void)d_ws; (void)ws_size;
    const float* f0 = (const float*)d_in[0];
    const float* f1 = (const float*)d_in[1];
    const float* f2 = (const float*)d_in[2];
    float* out = (float*)d_out;
    // 4 pairs * 2 batch * 128 rows * 8 x-tiles = 8192 waves, one wave32 each.
    lcs_wmma_kernel<<<8192, 32, 0, stream>>>(f0, f1, f2, out);
}